// GAT_24601572671662
// MI455X (gfx1250) — compile-verified
//
#include <hip/hip_runtime.h>
#include <math.h>

typedef __attribute__((ext_vector_type(2))) float v2f;
typedef __attribute__((ext_vector_type(8))) float v8f;

#define NEG_SLOPE 0.2f
#define LDS_STRIDE 132   // 128 + 4 pad: conflict-free b64 A-fragment reads

// ---------------------------------------------------------------------------
// WMMA GEMM:  Y[N x C] = X[N x K] * W[K x C]      (K = 128, C = 128 or 32)
// One wave per 16x16 output tile; block = one 16-row M-tile x all C/16 tiles.
// The 16xK A slab is staged once in LDS (shared by all waves in the block);
// A fragments come from ds_load_b64, B fragments from global (W is read
// exactly once per wave and is L2-resident).
// A-frag (16x4): lanes 0-15 rows M=0..15 holding K={0,1}; lanes 16-31 K={2,3}.
// C/D (16x16): VGPR i -> row i + 8*(lane>>4), col = lane&15.
// ---------------------------------------------------------------------------
__global__ void gat_gemm_wmma(const float* __restrict__ X,
                              const float* __restrict__ W,
                              float* __restrict__ Y,
                              int N, int K, int C) {
    __shared__ float sA[16 * LDS_STRIDE];

    const int lane = threadIdx.x & 31;
    const int wave = threadIdx.x >> 5;
    const int row0 = blockIdx.x * 16;
    const int col0 = wave * 16;
    const int m    = lane & 15;
    const int khi  = lane >> 4;          // 0 or 1: which K-pair this lane holds
    const int col  = col0 + m;           // B/D column for this lane

    // Cooperative load of the 16 x K A slab into LDS (float4 granularity).
    const int nvec = 16 * (K / 4);
    for (int i4 = threadIdx.x; i4 < nvec; i4 += blockDim.x) {
        const int r  = i4 / (K / 4);
        const int c4 = (i4 % (K / 4)) * 4;
        float4 v = *(const float4*)(X + (size_t)(row0 + r) * K + c4);
        float* p = &sA[r * LDS_STRIDE + c4];
        p[0] = v.x; p[1] = v.y; p[2] = v.z; p[3] = v.w;
    }
    __syncthreads();

    v8f acc = {};
    for (int kk = 0; kk < K; kk += 4) {
        const int ka = kk + 2 * khi;
        // A fragment from LDS: two consecutive K elements of row m
        v2f a = *(const v2f*)(&sA[m * LDS_STRIDE + ka]);
        // B fragment: W[ka, col], W[ka+1, col]
        v2f b;
        b.x = W[(size_t)ka * C + col];
        b.y = W[(size_t)(ka + 1) * C + col];
        acc = __builtin_amdgcn_wmma_f32_16x16x4_f32(
            /*neg_a=*/false, a, /*neg_b=*/false, b,
            /*c_mod=*/(short)0, acc, /*reuse_a=*/false, /*reuse_b=*/false);
    }
    // Store D tile
    #pragma unroll
    for (int i = 0; i < 8; ++i) {
        Y[(size_t)(row0 + i + 8 * khi) * C + col0 + m] = acc[i];
    }
}

// el[n,h] = sum_f feat[n,h,f]*a_src[h,f] ;  er likewise with a_dst
__global__ void gat_attn_coeff(const float* __restrict__ feat,
                               const float* __restrict__ a_s,
                               const float* __restrict__ a_d,
                               float* __restrict__ el, float* __restrict__ er,
                               int N, int H, int F) {
    int idx = blockIdx.x * blockDim.x + threadIdx.x;
    if (idx >= N * H) return;
    int h = idx % H;
    int n = idx / H;
    const float* fp = feat + ((size_t)n * H + h) * F;
    float sl = 0.f, sr = 0.f;
    for (int f = 0; f < F; ++f) {
        float v = fp[f];
        sl += v * a_s[h * F + f];
        sr += v * a_d[h * F + f];
    }
    el[idx] = sl;
    er[idx] = sr;
}

__global__ void gat_fill_f32(float* __restrict__ p, float v, int n) {
    int i = blockIdx.x * blockDim.x + threadIdx.x;
    if (i < n) p[i] = v;
}

__global__ void gat_init_bias(float* __restrict__ out, const float* __restrict__ b,
                              int N, int C) {
    int i = blockIdx.x * blockDim.x + threadIdx.x;
    if (i < N * C) out[i] = b[i % C];
}

// float atomic max via signed/unsigned integer atomics (valid with -inf init)
__device__ __forceinline__ void atomicMaxF(float* addr, float val) {
    if (val >= 0.f) atomicMax((int*)addr, __float_as_int(val));
    else            atomicMin((unsigned int*)addr, __float_as_uint(val));
}

// e = leaky_relu(el[src]+er[dst]); stash e; segment max into emax[dst]
__global__ void gat_edge_logits(const int* __restrict__ src, const int* __restrict__ dst,
                                const float* __restrict__ el, const float* __restrict__ er,
                                float* __restrict__ ee, float* __restrict__ emax,
                                int E, int H) {
    int idx = blockIdx.x * blockDim.x + threadIdx.x;
    if (idx >= E * H) return;
    int h = idx % H;
    int e = idx / H;
    int s = src[e], d = dst[e];
    float v = el[s * H + h] + er[d * H + h];
    v = (v > 0.f) ? v : NEG_SLOPE * v;
    ee[idx] = v;
    atomicMaxF(emax + d * H + h, v);
}

// ee = exp(e - emax[dst]); segment sum into denom[dst]
__global__ void gat_edge_exp(const int* __restrict__ dst,
                             float* __restrict__ ee, const float* __restrict__ emax,
                             float* __restrict__ denom, int E, int H) {
    int idx = blockIdx.x * blockDim.x + threadIdx.x;
    if (idx >= E * H) return;
    int h = idx % H;
    int e = idx / H;
    int d = dst[e];
    float v = __expf(ee[idx] - emax[d * H + h]);
    ee[idx] = v;
    atomicAdd(denom + d * H + h, v);
}

// alpha = ee / denom[dst]  (done once per (edge, head))
__global__ void gat_edge_norm(const int* __restrict__ dst,
                              float* __restrict__ ee, const float* __restrict__ denom,
                              int E, int H) {
    int idx = blockIdx.x * blockDim.x + threadIdx.x;
    if (idx >= E * H) return;
    int h = idx % H;
    int e = idx / H;
    ee[idx] = ee[idx] / denom[dst[e] * H + h];
}

// out[dst, c..c+3] += feat[src, c..c+3] * alpha[e,h]   (float4 gather, 4 atomics)
__global__ void gat_edge_scatter4(const int* __restrict__ src, const int* __restrict__ dst,
                                  const float* __restrict__ feat,
                                  const float* __restrict__ alpha,
                                  float* __restrict__ out, int E, int H, int F) {
    const int C  = H * F;
    const int C4 = C / 4;
    int idx = blockIdx.x * blockDim.x + threadIdx.x;
    if (idx >= E * C4) return;
    int e = idx / C4;
    int c = (idx % C4) * 4;
    int h = c / F;               // F multiple of 4 -> all 4 lanes same head
    int s = src[e], d = dst[e];
    float a = alpha[e * H + h];
    float4 f4 = *(const float4*)(feat + (size_t)s * C + c);
    float* op = out + (size_t)d * C + c;
    atomicAdd(op + 0, f4.x * a);
    atomicAdd(op + 1, f4.y * a);
    atomicAdd(op + 2, f4.z * a);
    atomicAdd(op + 3, f4.w * a);
}

__global__ void gat_elu_inplace(float* __restrict__ x, int n) {
    int i = blockIdx.x * blockDim.x + threadIdx.x;
    if (i >= n) return;
    float v = x[i];
    x[i] = (v > 0.f) ? v : (__expf(v) - 1.f);
}

// ---------------------------------------------------------------------------
extern "C" void kernel_launch(void* const* d_in, const int* in_sizes, int n_in,
                              void* d_out, int out_size, void* d_ws, size_t ws_size,
                              hipStream_t stream) {
    const float* features = (const float*)d_in[0];
    const int*   src      = (const int*)d_in[1];
    const int*   dst      = (const int*)d_in[2];
    const float* W1  = (const float*)d_in[3];
    const float* a1s = (const float*)d_in[4];
    const float* a1d = (const float*)d_in[5];
    const float* b1  = (const float*)d_in[6];
    const float* W2  = (const float*)d_in[7];
    const float* a2s = (const float*)d_in[8];
    const float* a2d = (const float*)d_in[9];
    const float* b2  = (const float*)d_in[10];
    const float* W3  = (const float*)d_in[11];
    const float* a3s = (const float*)d_in[12];
    const float* a3d = (const float*)d_in[13];
    const float* b3  = (const float*)d_in[14];

    const int IN = 128;                    // input feature dim (all layers)
    const int N  = in_sizes[0] / IN;       // 50000
    const int E  = in_sizes[1];            // 800000
    const int HMAX = 4;

    // workspace carve
    float* A    = (float*)d_ws;            // N*128 : layer input / aggregation
    float* FT   = A    + (size_t)N * 128;  // N*128 : feat = x @ W
    float* EE   = FT   + (size_t)N * 128;  // E*HMAX: edge logits / exp / alpha
    float* el   = EE   + (size_t)E * HMAX; // N*HMAX
    float* er   = el   + (size_t)N * HMAX;
    float* emax = er   + (size_t)N * HMAX;
    float* den  = emax + (size_t)N * HMAX;

    const int T = 256;
    auto blocks = [&](int n) { return (n + T - 1) / T; };

    auto run_layer = [&](const float* x, const float* Wm, const float* as_,
                         const float* ad_, const float* bias, float* featbuf,
                         float* aggbuf, int H, int F) {
        const int C  = H * F;
        const int NH = N * H;
        // GEMM via WMMA: N/16 tiles of rows, (C/16) waves per block over columns
        dim3 gg(N / 16);
        dim3 gb((C / 16) * 32);
        gat_gemm_wmma<<<gg, gb, 0, stream>>>(x, Wm, featbuf, N, IN, C);
        gat_attn_coeff<<<blocks(NH), T, 0, stream>>>(featbuf, as_, ad_, el, er, N, H, F);
        gat_fill_f32<<<blocks(NH), T, 0, stream>>>(emax, -INFINITY, NH);
        gat_fill_f32<<<blocks(NH), T, 0, stream>>>(den, 0.f, NH);
        gat_init_bias<<<blocks(N * C), T, 0, stream>>>(aggbuf, bias, N, C);
        gat_edge_logits<<<blocks(E * H), T, 0, stream>>>(src, dst, el, er, EE, emax, E, H);
        gat_edge_exp<<<blocks(E * H), T, 0, stream>>>(dst, EE, emax, den, E, H);
        gat_edge_norm<<<blocks(E * H), T, 0, stream>>>(dst, EE, den, E, H);
        gat_edge_scatter4<<<blocks(E * (C / 4)), T, 0, stream>>>(src, dst, featbuf, EE,
                                                                 aggbuf, E, H, F);
    };

    // Layer 1: features -> A, then ELU in place
    run_layer(features, W1, a1s, a1d, b1, FT, A, 4, 32);
    gat_elu_inplace<<<blocks(N * 128), T, 0, stream>>>(A, N * 128);

    // Layer 2: A -> (feat FT) -> A (A is free after the GEMM), then ELU
    run_layer(A, W2, a2s, a2d, b2, FT, A, 4, 32);
    gat_elu_inplace<<<blocks(N * 128), T, 0, stream>>>(A, N * 128);

    // Layer 3: H=1 -> mean over heads is identity; aggregate straight into d_out
    run_layer(A, W3, a3s, a3d, b3, FT, (float*)d_out, 1, 32);
}